// GraphAttention_17892833755267
// MI455X (gfx1250) — compile-verified
//
#include <hip/hip_runtime.h>

// ---------------------------------------------------------------------------
// GAT forward for MI455X (gfx1250, wave32).
//  K1: projections via V_WMMA_F32_16X16X4_F32 (fp32-exact matrix path)
//  K2: per-edge score + segment-max via ordered-key uint atomicMax
//  K3: exp + segment-sum (denominator)
//  K4: normalize + head-mean-folded scatter-add into final output
// ---------------------------------------------------------------------------

#define N_NODES   100000
#define N_EDGES   1600000
#define IN_F      256
#define H_HEADS   4
#define D_FEAT    32
#define HD        128          // H_HEADS * D_FEAT
#define NEG_SLOPE 0.2f

typedef __attribute__((ext_vector_type(2))) float v2f;
typedef __attribute__((ext_vector_type(8))) float v8f;

// Monotonic total-order key for f32 so we can use integer atomicMax.
__device__ __forceinline__ unsigned int f32_key(float f) {
    unsigned int u = __float_as_uint(f);
    return (u & 0x80000000u) ? ~u : (u | 0x80000000u);
}
__device__ __forceinline__ float key_f32(unsigned int k) {
    return (k & 0x80000000u) ? __uint_as_float(k ^ 0x80000000u)
                             : __uint_as_float(~k);
}
#define KEY_NEG_FLT_MAX 0x00800000u   // f32_key(-FLT_MAX)

// ---- K0: zero/−inf initialization (graph-replay safe: runs every call) ----
__global__ void gat_init_kernel(unsigned int* __restrict__ smax_key,
                                float* __restrict__ denom,
                                float* __restrict__ out0) {
    int i = blockIdx.x * blockDim.x + threadIdx.x;      // exact: N*32 threads
    out0[i] = 0.0f;
    if (i < N_NODES * H_HEADS) {
        smax_key[i] = KEY_NEG_FLT_MAX;
        denom[i]    = 0.0f;
    }
}

// ---- K1: el = x@w_src + b_src ; er = x@w_dst + b_dst via f32 WMMA ---------
// One wave per 16x16 output tile. Combined 256-wide N space: n-tiles 0..7 use
// w_src->el, 8..15 use w_dst->er. K loop: 64 x wmma_f32_16x16x4_f32.
// A layout (16x4 f32): lane<16 holds M=lane, {K0,K1}; lane>=16 holds {K2,K3}.
// B layout symmetric: N = lane&15 striped, K split by half-wave.
__global__ void gat_proj_wmma_kernel(const float* __restrict__ x,
                                     const float* __restrict__ w_src,
                                     const float* __restrict__ b_src,
                                     const float* __restrict__ w_dst,
                                     const float* __restrict__ b_dst,
                                     float* __restrict__ el,
                                     float* __restrict__ er) {
    const int wave = threadIdx.x >> 5;
    const int lane = threadIdx.x & 31;
    const int tile = blockIdx.x * 8 + wave;   // 0 .. 100000 (6250 Mtiles x 16)
    const int mt   = tile >> 4;
    const int nt   = tile & 15;

    const bool use_dst = (nt >= 8);           // uniform within wave
    const float* __restrict__ w   = use_dst ? w_dst : w_src;
    const float* __restrict__ bia = use_dst ? b_dst : b_src;
    float* __restrict__ outp      = use_dst ? er : el;

    const int col  = ((nt & 7) << 4) + (lane & 15);   // 0..127 within el/er
    const int half = lane >> 4;                       // 0: K0/K1, 1: K2/K3
    const int arow = (mt << 4) + (lane & 15);

    const float* __restrict__ xrow = x + (size_t)arow * IN_F + half * 2;
    const float* __restrict__ wcol = w + (size_t)(half * 2) * HD + col;

    v8f acc = {};
    #pragma unroll 4
    for (int k = 0; k < IN_F; k += 4) {
        v2f a = *(const v2f*)(xrow + k);              // A[M=arow, k+2h .. +1]
        v2f b;
        b.x = wcol[(size_t)k * HD];                   // B[k+2h,   col]
        b.y = wcol[(size_t)k * HD + HD];              // B[k+2h+1, col]
        acc = __builtin_amdgcn_wmma_f32_16x16x4_f32(
            /*neg_a=*/false, a, /*neg_b=*/false, b,
            /*c_mod=*/(short)0, acc, /*reuse_a=*/false, /*reuse_b=*/false);
    }

    const float bv = bia[col];
    #pragma unroll
    for (int r = 0; r < 8; ++r) {
        // C/D layout: VGPR r -> M = r (lanes 0-15) / r+8 (lanes 16-31)
        const int row = (mt << 4) + r + (half << 3);
        outp[(size_t)row * HD + col] = acc[r] + bv;
    }
}

// ---- K2: edge scores s[e,h] + running segment max --------------------------
// One wave per edge: 128 consecutive floats per gather (512B coalesced).
__global__ void gat_edge_score_kernel(const float* __restrict__ el,
                                      const float* __restrict__ er,
                                      const float* __restrict__ attn,
                                      const int* __restrict__ src,
                                      const int* __restrict__ dst,
                                      float* __restrict__ sbuf,
                                      unsigned int* __restrict__ smax_key) {
    const int e    = blockIdx.x * 8 + (threadIdx.x >> 5);
    const int lane = threadIdx.x & 31;
    const int s = src[e];
    const int d = dst[e];

    const float4 vl = *(const float4*)(el   + (size_t)s * HD + lane * 4);
    const float4 vr = *(const float4*)(er   + (size_t)d * HD + lane * 4);
    const float4 at = *(const float4*)(attn + lane * 4);

    float t, sum = 0.0f;
    t = vl.x + vr.x; sum += at.x * (t > 0.0f ? t : NEG_SLOPE * t);
    t = vl.y + vr.y; sum += at.y * (t > 0.0f ? t : NEG_SLOPE * t);
    t = vl.z + vr.z; sum += at.z * (t > 0.0f ? t : NEG_SLOPE * t);
    t = vl.w + vr.w; sum += at.w * (t > 0.0f ? t : NEG_SLOPE * t);

    // reduce over the 8 lanes of each head (wave32 shuffles)
    sum += __shfl_xor(sum, 1);
    sum += __shfl_xor(sum, 2);
    sum += __shfl_xor(sum, 4);

    if ((lane & 7) == 0) {
        const int h = lane >> 3;
        sbuf[(size_t)e * H_HEADS + h] = sum;
        atomicMax(smax_key + (size_t)d * H_HEADS + h, f32_key(sum));
    }
}

// ---- K3: ex = exp(s - smax[dst]) (in place) + denominator ------------------
__global__ void gat_softmax_num_kernel(const int* __restrict__ dst,
                                       float* __restrict__ sbuf,
                                       const unsigned int* __restrict__ smax_key,
                                       float* __restrict__ denom) {
    const int i = blockIdx.x * blockDim.x + threadIdx.x;  // exact: E*H threads
    const int e = i >> 2;
    const int h = i & 3;
    const int d = dst[e];
    const float m  = key_f32(smax_key[d * H_HEADS + h]);
    const float ex = __expf(sbuf[i] - m);
    sbuf[i] = ex;                                          // overwrite s -> ex
    atomicAdd(denom + d * H_HEADS + h, ex);
}

// ---- K4: a = ex/denom; out1[e]=mean_h a; out0 += mean_h(el[src]*a) ---------
// Head-mean folded into the scatter: cross-head lane reduction leaves only
// 32 f32 atomicAdds/edge landing in the final (N,32) output (L2-resident).
__global__ void gat_aggregate_kernel(const float* __restrict__ el,
                                     const int* __restrict__ src,
                                     const int* __restrict__ dst,
                                     const float* __restrict__ exbuf,
                                     const float* __restrict__ denom,
                                     float* __restrict__ out0,
                                     float* __restrict__ out1) {
    const int e    = blockIdx.x * 8 + (threadIdx.x >> 5);
    const int lane = threadIdx.x & 31;
    const int s = src[e];
    const int d = dst[e];
    const int h = lane >> 3;

    const float a = exbuf[(size_t)e * H_HEADS + h] / denom[d * H_HEADS + h];

    if (lane == 0) {
        float asum = 0.0f;
        #pragma unroll
        for (int hh = 0; hh < H_HEADS; ++hh)
            asum += exbuf[(size_t)e * H_HEADS + hh] / denom[d * H_HEADS + hh];
        out1[e] = 0.25f * asum;
    }

    const float4 vl = *(const float4*)(el + (size_t)s * HD + lane * 4);
    const float sc = a * 0.25f;                       // fold mean over heads
    float4 c;
    c.x = vl.x * sc; c.y = vl.y * sc; c.z = vl.z * sc; c.w = vl.w * sc;

    // lanes i, i+8, i+16, i+24 cover the same d-range for the 4 heads
    c.x += __shfl_xor(c.x, 8);  c.y += __shfl_xor(c.y, 8);
    c.z += __shfl_xor(c.z, 8);  c.w += __shfl_xor(c.w, 8);
    c.x += __shfl_xor(c.x, 16); c.y += __shfl_xor(c.y, 16);
    c.z += __shfl_xor(c.z, 16); c.w += __shfl_xor(c.w, 16);

    if (lane < 8) {
        float* op = out0 + (size_t)d * D_FEAT + lane * 4;
        atomicAdd(op + 0, c.x);
        atomicAdd(op + 1, c.y);
        atomicAdd(op + 2, c.z);
        atomicAdd(op + 3, c.w);
    }
}

// ---------------------------------------------------------------------------
extern "C" void kernel_launch(void* const* d_in, const int* in_sizes, int n_in,
                              void* d_out, int out_size, void* d_ws, size_t ws_size,
                              hipStream_t stream) {
    (void)in_sizes; (void)n_in; (void)out_size; (void)ws_size;

    const float* x     = (const float*)d_in[0];
    const float* w_src = (const float*)d_in[1];
    const float* b_src = (const float*)d_in[2];
    const float* w_dst = (const float*)d_in[3];
    const float* b_dst = (const float*)d_in[4];
    const float* attn  = (const float*)d_in[5];
    const int*   src   = (const int*)d_in[6];
    const int*   dst   = (const int*)d_in[7];

    float* out0 = (float*)d_out;                        // (N, 32) mean feats
    float* out1 = out0 + (size_t)N_NODES * D_FEAT;      // (E, 1)  mean attn

    // workspace carve-up (~131 MB)
    float* el            = (float*)d_ws;                            // N*128
    float* er            = el + (size_t)N_NODES * HD;               // N*128
    float* sbuf          = er + (size_t)N_NODES * HD;               // E*4
    unsigned int* smaxk  = (unsigned int*)(sbuf + (size_t)N_EDGES * H_HEADS); // N*4
    float* denom         = (float*)(smaxk + (size_t)N_NODES * H_HEADS);       // N*4

    // K0: init (N*32 threads exactly)
    gat_init_kernel<<<N_NODES * D_FEAT / 256, 256, 0, stream>>>(smaxk, denom, out0);

    // K1: 6250 Mtiles * 16 Ntiles = 100000 waves, 8 waves/block
    gat_proj_wmma_kernel<<<(N_NODES / 16) * 16 / 8, 256, 0, stream>>>(
        x, w_src, b_src, w_dst, b_dst, el, er);

    // K2: one wave per edge
    gat_edge_score_kernel<<<N_EDGES / 8, 256, 0, stream>>>(
        el, er, attn, src, dst, sbuf, smaxk);

    // K3: one thread per (edge, head)
    gat_softmax_num_kernel<<<N_EDGES * H_HEADS / 256, 256, 0, stream>>>(
        dst, sbuf, smaxk, denom);

    // K4: one wave per edge
    gat_aggregate_kernel<<<N_EDGES / 8, 256, 0, stream>>>(
        el, src, dst, sbuf, denom, out0, out1);
}